// GatedCrossAttentionFuse_82197084111476
// MI455X (gfx1250) — compile-verified
//
#include <hip/hip_runtime.h>
#include <math.h>

// ---------------------------------------------------------------------------
// GatedCrossAttentionFuse for MI455X (gfx1250, wave32, WMMA)
// B=4, C_LIDAR=128, H=W=256, C_CAM=256, HEADS=4, DIM_HEAD=32, HD=128, N_TOK=8192
// ---------------------------------------------------------------------------

#define NTOK 8192
#define HD_   128
#define CCAM  256
#define HW_   65536   // 256*256
#define HDIM  256
#define WDIM  256

typedef __attribute__((ext_vector_type(2))) float v2f;
typedef __attribute__((ext_vector_type(8))) float v8f;

__device__ __forceinline__ int clamp255(int x) {
    return x < 0 ? 0 : (x > 255 ? 255 : x);
}

// ---------------------------------------------------------------------------
// out[b,c,h,w] = lidar[b,c,h,w] + range_alpha * out_bias[c]   (dense base)
// ---------------------------------------------------------------------------
__global__ __launch_bounds__(256) void base_kernel(
    const float* __restrict__ lidar, const float* __restrict__ out_bias,
    const float* __restrict__ alpha_p, float* __restrict__ out)
{
    size_t tid = (size_t)blockIdx.x * 256 + threadIdx.x;   // over B*128*HW
    int c = (int)((tid >> 16) & 127);                       // (tid / 65536) % 128
    out[tid] = lidar[tid] + out_bias[c] * alpha_p[0];
}

// ---------------------------------------------------------------------------
// G[t, c] = lidar_b[c, i_t, j_t]   (gather token features from BEV)
// ---------------------------------------------------------------------------
__global__ __launch_bounds__(256) void gather_kernel(
    const float* __restrict__ lidar_b, const int* __restrict__ ind_b,
    float* __restrict__ G)
{
    int tid = blockIdx.x * 256 + threadIdx.x;   // over NTOK*128
    int t = tid >> 7;
    int c = tid & 127;
    int i = clamp255(ind_b[t * 2 + 0]);
    int j = clamp255(ind_b[t * 2 + 1]);
    G[tid] = lidar_b[(size_t)c * HW_ + i * WDIM + j];
}

// ---------------------------------------------------------------------------
// Y[m, n] = sum_k A[m,k] * W[n,k] (+ bias[n])      via V_WMMA_F32_16X16X4_F32
// A: [NTOK, Cin] row-major, W: [128, Cin] row-major, Y: [NTOK, 128].
// One wave per 16-token strip; 8 N-tiles of 16 -> 8 v8f accumulators.
// A-fragment: lane holds row (lane&15), K = kb + (lane>>4)*2 .. +1  (float2)
// B-fragment: lane holds col (lane&15), same K pair -> contiguous in W's row.
// ---------------------------------------------------------------------------
__global__ __launch_bounds__(256) void gemm_xwT_kernel(
    const float* __restrict__ A, const float* __restrict__ W,
    const float* __restrict__ bias, float* __restrict__ Y,
    int Cin, int has_bias)
{
    const int lane  = threadIdx.x & 31;
    const int wave  = threadIdx.x >> 5;
    const int m0    = (blockIdx.x * 8 + wave) * 16;   // token-tile base row
    const int nrow  = lane & 15;                      // row/col within tile
    const int kpair = (lane >> 4) << 1;               // 0 or 2
    const int half  = lane >> 4;                      // 0 or 1

    v8f acc[8] = {};

    const float* arow = A + (size_t)(m0 + nrow) * Cin + kpair;

    const float* wrow[8];
#pragma unroll
    for (int nt = 0; nt < 8; ++nt)
        wrow[nt] = W + (size_t)(nt * 16 + nrow) * Cin + kpair;

    for (int kb = 0; kb < Cin; kb += 4) {
        v2f afrag = *(const v2f*)(arow + kb);
#pragma unroll
        for (int nt = 0; nt < 8; ++nt) {
            v2f bfrag = *(const v2f*)(wrow[nt] + kb);
            acc[nt] = __builtin_amdgcn_wmma_f32_16x16x4_f32(
                false, afrag, false, bfrag, (short)0, acc[nt], false, false);
        }
    }

    // Epilogue: VGPR e of lane l holds C element (M = e + (l>>4)*8, N = l&15)
#pragma unroll
    for (int nt = 0; nt < 8; ++nt) {
        int n = nt * 16 + nrow;
        float bv = has_bias ? bias[n] : 0.0f;
#pragma unroll
        for (int e = 0; e < 8; ++e) {
            int m = m0 + e + half * 8;
            Y[(size_t)m * HD_ + n] = acc[nt][e] + bv;
        }
    }
}

// ---------------------------------------------------------------------------
// logits[h, t] = scale * dot32(Q[t, h*32:], K[t, h*32:])
// ---------------------------------------------------------------------------
__global__ __launch_bounds__(256) void logits_kernel(
    const float* __restrict__ Q, const float* __restrict__ K,
    float* __restrict__ logits)
{
    int tid = blockIdx.x * 256 + threadIdx.x;   // over NTOK*4
    int t = tid >> 2;
    int h = tid & 3;
    const float* q = Q + (size_t)t * HD_ + h * 32;
    const float* k = K + (size_t)t * HD_ + h * 32;
    float s = 0.0f;
#pragma unroll
    for (int d = 0; d < 32; ++d) s += q[d] * k[d];
    logits[h * NTOK + t] = s * 0.17677669529663687f;   // 1/sqrt(32)
}

// ---------------------------------------------------------------------------
// attn[h, t] = softmax_t(logits[h, :])[t] * gw[t]     (one block per head)
// ---------------------------------------------------------------------------
__global__ __launch_bounds__(256) void softmax_kernel(
    const float* __restrict__ logits, const float* __restrict__ gw,
    float* __restrict__ attn)
{
    __shared__ float sdata[256];
    const int h = blockIdx.x;
    const int tid = threadIdx.x;
    const float* L = logits + h * NTOK;
    float* Aout = attn + h * NTOK;

    float m = -INFINITY;
    for (int t = tid; t < NTOK; t += 256) m = fmaxf(m, L[t]);
    sdata[tid] = m;
    __syncthreads();
    for (int s = 128; s > 0; s >>= 1) {
        if (tid < s) sdata[tid] = fmaxf(sdata[tid], sdata[tid + s]);
        __syncthreads();
    }
    float mx = sdata[0];
    __syncthreads();

    float sum = 0.0f;
    for (int t = tid; t < NTOK; t += 256) {
        float e = __expf(L[t] - mx);
        Aout[t] = e;
        sum += e;
    }
    sdata[tid] = sum;
    __syncthreads();
    for (int s = 128; s > 0; s >>= 1) {
        if (tid < s) sdata[tid] += sdata[tid + s];
        __syncthreads();
    }
    float inv = 1.0f / sdata[0];

    for (int t = tid; t < NTOK; t += 256) Aout[t] = Aout[t] * inv * gw[t];
}

// ---------------------------------------------------------------------------
// V[t, c] *= attn[c>>5, t]   (in place: V becomes "fused")
// ---------------------------------------------------------------------------
__global__ __launch_bounds__(256) void scale_v_kernel(
    float* __restrict__ V, const float* __restrict__ attn)
{
    int tid = blockIdx.x * 256 + threadIdx.x;   // over NTOK*128
    int t = tid >> 7;
    int c = tid & 127;
    V[tid] *= attn[(c >> 5) * NTOK + t];
}

// ---------------------------------------------------------------------------
// out_b[c, i_t, j_t] += range_alpha * contrib[t, c]   (atomic scatter-add)
// ---------------------------------------------------------------------------
__global__ __launch_bounds__(256) void scatter_kernel(
    const float* __restrict__ contrib, const int* __restrict__ ind_b,
    const float* __restrict__ alpha_p, float* __restrict__ out_b)
{
    int tid = blockIdx.x * 256 + threadIdx.x;   // over NTOK*128
    int t = tid >> 7;
    int c = tid & 127;
    int i = clamp255(ind_b[t * 2 + 0]);
    int j = clamp255(ind_b[t * 2 + 1]);
    atomicAdd(&out_b[(size_t)c * HW_ + i * WDIM + j], contrib[tid] * alpha_p[0]);
}

// ---------------------------------------------------------------------------
extern "C" void kernel_launch(void* const* d_in, const int* in_sizes, int n_in,
                              void* d_out, int out_size, void* d_ws, size_t ws_size,
                              hipStream_t stream)
{
    const float* lidar  = (const float*)d_in[0];   // [4,128,256,256]
    const float* tok    = (const float*)d_in[1];   // [4,8192,256]
    const int*   ind    = (const int*)d_in[2];     // [4,8192,2]
    const float* gw     = (const float*)d_in[3];   // [4,8192]
    const float* alpha  = (const float*)d_in[4];   // scalar
    const float* q_w    = (const float*)d_in[5];   // [128,128]
    const float* q_b    = (const float*)d_in[6];   // [128]
    const float* k_w    = (const float*)d_in[7];   // [128,256]
    const float* v_w    = (const float*)d_in[8];   // [128,256]
    const float* out_w  = (const float*)d_in[9];   // [128,128]
    const float* out_b  = (const float*)d_in[10];  // [128]
    float* out = (float*)d_out;

    // Workspace layout (reused per batch; ~21.2 MB)
    float* G     = (float*)d_ws;          // [NTOK,128] gathered lidar feats
    float* Q     = G   + NTOK * HD_;      // [NTOK,128]
    float* K     = Q   + NTOK * HD_;      // [NTOK,128]
    float* V     = K   + NTOK * HD_;      // [NTOK,128] -> fused (in place)
    float* Ctr   = V   + NTOK * HD_;      // [NTOK,128] out_w projection
    float* logit = Ctr + NTOK * HD_;      // [4,NTOK]
    float* attn  = logit + 4 * NTOK;      // [4,NTOK]

    // Dense base: out = lidar + alpha * out_bias  (covers all 4 batches)
    base_kernel<<<(4 * 128 * HW_) / 256, 256, 0, stream>>>(lidar, out_b, alpha, out);

    const int tc_blocks   = (NTOK * HD_) / 256;   // 4096
    const int gemm_blocks = NTOK / (16 * 8);      // 64 (8 waves/block)

    for (int b = 0; b < 4; ++b) {
        const float* lidar_b = lidar + (size_t)b * 128 * HW_;
        const float* tok_b   = tok   + (size_t)b * NTOK * CCAM;
        const int*   ind_b   = ind   + (size_t)b * NTOK * 2;
        const float* gw_b    = gw    + (size_t)b * NTOK;
        float*       out_bp  = out   + (size_t)b * 128 * HW_;

        gather_kernel<<<tc_blocks, 256, 0, stream>>>(lidar_b, ind_b, G);

        // Q = G @ q_w^T + q_bias ;  K = tok @ k_w^T ;  V = tok @ v_w^T
        gemm_xwT_kernel<<<gemm_blocks, 256, 0, stream>>>(G,     q_w, q_b,     Q, 128, 1);
        gemm_xwT_kernel<<<gemm_blocks, 256, 0, stream>>>(tok_b, k_w, nullptr, K, CCAM, 0);
        gemm_xwT_kernel<<<gemm_blocks, 256, 0, stream>>>(tok_b, v_w, nullptr, V, CCAM, 0);

        logits_kernel<<<(NTOK * 4) / 256, 256, 0, stream>>>(Q, K, logit);
        softmax_kernel<<<4, 256, 0, stream>>>(logit, gw_b, attn);
        scale_v_kernel<<<tc_blocks, 256, 0, stream>>>(V, attn);

        // contrib = fused @ out_w^T, then scatter-add * alpha into out
        gemm_xwT_kernel<<<gemm_blocks, 256, 0, stream>>>(V, out_w, nullptr, Ctr, 128, 0);
        scatter_kernel<<<tc_blocks, 256, 0, stream>>>(Ctr, ind_b, alpha, out_bp);
    }
}